// ComputeAngleInput_81827716923455
// MI455X (gfx1250) — compile-verified
//
#include <hip/hip_runtime.h>
#include <hip/hip_bf16.h>

typedef __attribute__((ext_vector_type(2))) float v2f;
typedef __attribute__((ext_vector_type(8))) float v8f;

#define N_NEIGH 16
#define N_FEAT  32
#define N_PAIR  240                  // 16*16 - 16
#define ROW_LEN 99                   // 3 + 3*32
#define TOT     (N_PAIR * ROW_LEN)   // 23760 floats per center

// Contiguous LDS table layout (float indices, relative to smem[0]):
#define OFF_D    0      // 16   : d_ij row
#define OFF_TJK  16     // 256  : normalized d_jk (16x16)
#define OFF_EMBI 272    // 32   : center embedding
#define OFF_EJD  304    // 512  : emb_j[f] / d_ij   (16x32)
#define SMEM_F   816

// Static element -> LDS-byte-offset map (same for every center).
__device__ __align__(16) unsigned short g_srcoff[TOT];

__global__ __launch_bounds__(256) void build_lut_kernel() {
    const int t = blockIdx.x * 256 + threadIdx.x;
    if (t >= TOT) return;
    const unsigned p  = (unsigned)t / 99u;
    const unsigned e  = (unsigned)t - p * 99u;
    const unsigned j  = p / 15u;
    const unsigned rr = p - j * 15u;
    const unsigned k  = rr + (rr >= j ? 1u : 0u);
    unsigned idx;
    if      (e == 0u) idx = OFF_D    + j;
    else if (e == 1u) idx = OFF_D    + k;
    else if (e == 2u) idx = OFF_TJK  + j * 16u + k;
    else if (e < 35u) idx = OFF_EMBI + (e - 3u);
    else if (e < 67u) idx = OFF_EJD  + j * 32u + (e - 35u);
    else              idx = OFF_EJD  + k * 32u + (e - 67u);
    g_srcoff[t] = (unsigned short)(idx * 4u);   // byte offset, max 3260
}

__global__ __launch_bounds__(256) void angdesc_wmma_kernel(
    const float* __restrict__ atoms_xyz,    // (N_ATOMS, 3)
    const float* __restrict__ embed_table,  // (N_TYPES, 32)
    const float* __restrict__ dist_ij,      // (C, 16)
    const int*   __restrict__ atom_types,   // (N_ATOMS,)
    const int*   __restrict__ atom_i_idx,   // (C,)
    const int*   __restrict__ atom_j_idx,   // (C, 16)
    float*       __restrict__ out,          // C*TOT ang_desc, then C centers
    int C)
{
    __shared__ float smem[SMEM_F];          // LUT-addressed region
    __shared__ float s_x[16], s_y[16], s_z[16], s_n2[16], s_invd[16];
    __shared__ int   s_jtype[16];

    const int tid = threadIdx.x;
    const int c   = blockIdx.x;

    // ---- Phase 1: per-neighbor gathers -----------------------------------
    if (tid < 16) {
        const int j = atom_j_idx[(size_t)c * N_NEIGH + tid];
        s_jtype[tid] = atom_types[j];
        const float x = atoms_xyz[3 * (size_t)j + 0];
        const float y = atoms_xyz[3 * (size_t)j + 1];
        const float z = atoms_xyz[3 * (size_t)j + 2];
        s_x[tid] = x; s_y[tid] = y; s_z[tid] = z;
        s_n2[tid] = x * x + y * y + z * z;
        const float d = dist_ij[(size_t)c * N_NEIGH + tid];
        smem[OFF_D + tid] = d;
        s_invd[tid] = 1.0f / d;
    }
    if (tid < 32) {
        const int ii = atom_i_idx[c];
        const int it = atom_types[ii];
        smem[OFF_EMBI + tid] = embed_table[it * N_FEAT + tid];
    }
    __syncthreads();

    // ---- Phase 2: neighbor embeddings, pre-divided by d_ij ---------------
    for (int idx = tid; idx < N_NEIGH * N_FEAT; idx += 256) {
        const int j = idx >> 5;
        const int f = idx & 31;
        smem[OFF_EJD + idx] = embed_table[s_jtype[j] * N_FEAT + f] * s_invd[j];
    }

    // ---- Phase 3: wave 0 computes the 16x16 Gram matrix via WMMA ---------
    // G = X * X^T, X (16x3) zero-padded to K=4.  With the CDNA5 16x4 f32 A
    // layout (lanes 0-15: K=0/1, lanes 16-31: K=2/3) and the mirrored B
    // layout, the A and B operands of a Gram matrix are the same registers.
    if (tid < 32) {
        const int  lane = tid;
        const bool hi   = lane >= 16;
        const int  l16  = lane & 15;

        v2f a;
        a.x = hi ? s_z[l16] : s_x[l16];   // K=0 (lo) / K=2 (hi)
        a.y = hi ? 0.0f     : s_y[l16];   // K=1 (lo) / K=3 pad (hi)

        v8f acc = {};
        acc = __builtin_amdgcn_wmma_f32_16x16x4_f32(
            /*neg_a=*/false, a, /*neg_b=*/false, a,
            /*c_mod=*/(short)0, acc, /*reuse_a=*/false, /*reuse_b=*/false);

        #pragma unroll
        for (int r = 0; r < 8; ++r) {
            const int m  = hi ? (r + 8) : r;   // row j
            const int nn = l16;                // col k
            const float dot = acc[r];
            const float d2  = s_n2[m] + s_n2[nn] - 2.0f * dot;
            const float djk = sqrtf(fmaxf(d2, 0.0f));
            const float dij = smem[OFF_D + m];
            const float dik = smem[OFF_D + nn];
            const float mx  = fmaxf(dij, dik);
            const float mn  = fminf(dij, dik);
            smem[OFF_TJK + m * 16 + nn] = (djk - mx + mn) / (2.0f * mn);
        }
    }
    __syncthreads();

    // ---- Phase 4: stream TOT floats per center via the static LUT --------
    // Per float4: one 8B LUT load (WGP$-resident), 4x ds_load_b32, one
    // global_store_b128.  No divisions / select chains in the hot loop.
    const size_t base = (size_t)c * TOT;
    const char* smem_b = reinterpret_cast<const char*>(smem);
    for (int t = tid * 4; t < TOT; t += 256 * 4) {
        const ushort4 o = *reinterpret_cast<const ushort4*>(&g_srcoff[t]);
        float4 v;
        v.x = *reinterpret_cast<const float*>(smem_b + o.x);
        v.y = *reinterpret_cast<const float*>(smem_b + o.y);
        v.z = *reinterpret_cast<const float*>(smem_b + o.z);
        v.w = *reinterpret_cast<const float*>(smem_b + o.w);
        *reinterpret_cast<float4*>(out + base + t) = v;
    }

    // ---- Phase 5: centers tuple member -----------------------------------
    if (tid == 0) {
        out[(size_t)C * TOT + c] = (float)atom_i_idx[c];
    }
}

extern "C" void kernel_launch(void* const* d_in, const int* in_sizes, int n_in,
                              void* d_out, int out_size, void* d_ws, size_t ws_size,
                              hipStream_t stream) {
    const float* atoms_xyz   = (const float*)d_in[0];
    const float* embed_table = (const float*)d_in[1];
    const float* dist_ij     = (const float*)d_in[2];
    const int*   atom_types  = (const int*)  d_in[3];
    const int*   atom_i_idx  = (const int*)  d_in[4];
    const int*   atom_j_idx  = (const int*)  d_in[5];
    float*       out         = (float*)d_out;

    const int C = in_sizes[2] / N_NEIGH;   // dist_ij has C*16 elements

    build_lut_kernel<<<dim3((TOT + 255) / 256), dim3(256), 0, stream>>>();
    angdesc_wmma_kernel<<<dim3(C), dim3(256), 0, stream>>>(
        atoms_xyz, embed_table, dist_ij, atom_types, atom_i_idx, atom_j_idx,
        out, C);
}